// SpectrumLoss_18279380812462
// MI455X (gfx1250) — compile-verified
//
#include <hip/hip_runtime.h>
#include <math.h>

// ---------------------------------------------------------------------------
// SpectrumLoss on MI455X (gfx1250): 2D DFT as F*X*F^T via V_WMMA_F32_16X16X4_F32
// x: [B=32, H=256, W=256, C=16] f32 -> scalar loss (f32)
// Round 2: 2x2 register blocking (32x32 super-tile per wave) to double the
// wmma-per-load ratio and give the scheduler 4/8 independent accumulators.
// ---------------------------------------------------------------------------

typedef __attribute__((ext_vector_type(2))) float v2f;
typedef __attribute__((ext_vector_type(8))) float v8f;

#define HH 256
#define BB 32
#define CC 16
#define NIMG (BB * CC)   /* 512 */
#define NBINS 128
#define NBINS1 129
#define CHUNK 16
#define NCHUNK (NIMG / CHUNK)

// workspace layout (float offsets)
#define OFF_FRE  0
#define OFF_FIM  (OFF_FRE + HH * HH)                    /* 65536  */
#define OFF_BSUM (OFF_FIM + HH * HH)                    /* 131072 */
#define OFF_CNT  (OFF_BSUM + NIMG * NBINS1)             /* 197120 (uint) */
#define OFF_LOSS (OFF_CNT + NBINS1)                     /* 197249 */
#define OFF_Y    (((OFF_LOSS + NIMG) + 127) & ~127)     /* 197888 */
// Y chunk: CHUNK * 2 * 256*256 floats = 2,097,152 -> total ws ~9.2 MB

__device__ __forceinline__ v8f wmma4(v2f a, v2f b, v8f c) {
  // D = A(16x4,f32) * B(4x16,f32) + C(16x16,f32)
  return __builtin_amdgcn_wmma_f32_16x16x4_f32(false, a, false, b, (short)0, c,
                                               false, false);
}

__device__ __forceinline__ int radial_bin(int ks, int js) {
  float fy = (float)(ks - 128) * (1.0f / 256.0f);
  float fx = (float)(js - 128) * (1.0f / 256.0f);
  float fr = sqrtf(fy * fy + fx * fx);
  int bin = (int)(fr * 256.0f);   // uniform edges: width 1/256, exact at lattice hits
  return bin > NBINS ? NBINS : bin;
}

// --- zero accumulators ------------------------------------------------------
__global__ void sl_init(float* ws) {
  int i = blockIdx.x * blockDim.x + threadIdx.x;
  int n = NIMG * NBINS1 + NBINS1;  // bin sums + counts
  if (i < n) ws[OFF_BSUM + i] = 0.0f;
}

// --- build DFT matrix (ortho-normalized) + radial bin counts ----------------
__global__ void sl_build(float* ws) {
  int id = blockIdx.x * blockDim.x + threadIdx.x;
  if (id >= HH * HH) return;
  int k = id >> 8, n = id & 255;
  int m = (k * n) & 255;                       // exact period reduction
  float ang = (float)m * (6.283185307179586f / 256.0f);
  ws[OFF_FRE + id] = cosf(ang) * 0.0625f;      // 1/sqrt(256) per stage
  ws[OFF_FIM + id] = -sinf(ang) * 0.0625f;
  // counts over the shifted grid (ks = id>>8, js = id&255)
  unsigned* cnt = (unsigned*)(ws + OFF_CNT);
  atomicAdd(&cnt[radial_bin(k, n)], 1u);
}

// --- stage 1: Y = F * X  (32x32 super-tile per wave) ------------------------
__global__ void __launch_bounds__(256) sl_stage1(const float* __restrict__ x,
                                                 float* __restrict__ ws,
                                                 int imgBase) {
  int lane = threadIdx.x & 31;
  int laneLo = lane & 15, laneHi = lane >> 4;
  int wave = (blockIdx.x * 256 + threadIdx.x) >> 5;  // 0 .. 1023
  int imgLocal = wave >> 6;                          // 64 super-tiles / image
  int superId = wave & 63;
  int kS = superId >> 3, wS = superId & 7;           // 8x8 grid of 32x32 tiles
  int g = imgBase + imgLocal;
  int b = g >> 4, c = g & 15;

  const float* Fre = ws + OFF_FRE;
  const float* Fim = ws + OFF_FIM;
  float* Yre = ws + OFF_Y + (size_t)imgLocal * 2 * HH * HH;
  float* Yim = Yre + HH * HH;
  const float* X = x + (size_t)b * HH * HH * CC + c;  // X[h][w] = X[(h*256+w)*16]

  int arow0 = kS * 32 + laneLo, arow1 = arow0 + 16;
  int bcol0 = wS * 32 + laneLo, bcol1 = bcol0 + 16;

  v8f accRe00 = {}, accRe01 = {}, accRe10 = {}, accRe11 = {};
  v8f accIm00 = {}, accIm01 = {}, accIm10 = {}, accIm11 = {};
  for (int h = 0; h < HH; h += 4) {
    int ac = h + 2 * laneHi;  // K-slot per ISA A/B 16x4 layout
    v2f aRe0 = {Fre[arow0 * HH + ac], Fre[arow0 * HH + ac + 1]};
    v2f aRe1 = {Fre[arow1 * HH + ac], Fre[arow1 * HH + ac + 1]};
    v2f aIm0 = {Fim[arow0 * HH + ac], Fim[arow0 * HH + ac + 1]};
    v2f aIm1 = {Fim[arow1 * HH + ac], Fim[arow1 * HH + ac + 1]};
    v2f bX0 = {X[((size_t)ac * HH + bcol0) * CC],
               X[((size_t)(ac + 1) * HH + bcol0) * CC]};
    v2f bX1 = {X[((size_t)ac * HH + bcol1) * CC],
               X[((size_t)(ac + 1) * HH + bcol1) * CC]};
    accRe00 = wmma4(aRe0, bX0, accRe00);
    accRe01 = wmma4(aRe0, bX1, accRe01);
    accRe10 = wmma4(aRe1, bX0, accRe10);
    accRe11 = wmma4(aRe1, bX1, accRe11);
    accIm00 = wmma4(aIm0, bX0, accIm00);
    accIm01 = wmma4(aIm0, bX1, accIm01);
    accIm10 = wmma4(aIm1, bX0, accIm10);
    accIm11 = wmma4(aIm1, bX1, accIm11);
  }
#pragma unroll
  for (int r = 0; r < 8; ++r) {
    int r0 = kS * 32 + r + 8 * laneHi;  // C/D layout: M = r (+8 for hi lanes)
    int r1 = r0 + 16;
    Yre[r0 * HH + bcol0] = accRe00[r];
    Yre[r0 * HH + bcol1] = accRe01[r];
    Yre[r1 * HH + bcol0] = accRe10[r];
    Yre[r1 * HH + bcol1] = accRe11[r];
    Yim[r0 * HH + bcol0] = accIm00[r];
    Yim[r0 * HH + bcol1] = accIm01[r];
    Yim[r1 * HH + bcol0] = accIm10[r];
    Yim[r1 * HH + bcol1] = accIm11[r];
  }
}

// --- stage 2: Z = Y * F^T (32x32 super-tile), power, fftshift, histogram ----
__global__ void __launch_bounds__(256) sl_stage2(float* __restrict__ ws,
                                                 int imgBase) {
  __shared__ float hist[NBINS1];
  int t = threadIdx.x;
  if (t < NBINS1) hist[t] = 0.0f;
  __syncthreads();

  int lane = t & 31;
  int laneLo = lane & 15, laneHi = lane >> 4;
  int imgLocal = blockIdx.x >> 3;                 // 8 blocks (64 waves) / image
  int superId = (blockIdx.x & 7) * 8 + (t >> 5);  // 8x8 grid of 32x32 tiles
  int kS = superId >> 3, jS = superId & 7;

  const float* Fre = ws + OFF_FRE;
  const float* Fim = ws + OFF_FIM;
  const float* Yre = ws + OFF_Y + (size_t)imgLocal * 2 * HH * HH;
  const float* Yim = Yre + HH * HH;

  int arow0 = kS * 32 + laneLo, arow1 = arow0 + 16;
  int jcol0 = jS * 32 + laneLo, jcol1 = jcol0 + 16;

  v8f zRe00 = {}, zRe01 = {}, zRe10 = {}, zRe11 = {};
  v8f zIm00 = {}, zIm01 = {}, zIm10 = {}, zIm11 = {};
  for (int w = 0; w < HH; w += 4) {
    int ac = w + 2 * laneHi;
    v2f aRe0 = {Yre[arow0 * HH + ac], Yre[arow0 * HH + ac + 1]};
    v2f aRe1 = {Yre[arow1 * HH + ac], Yre[arow1 * HH + ac + 1]};
    v2f aIm0 = {Yim[arow0 * HH + ac], Yim[arow0 * HH + ac + 1]};
    v2f aIm1 = {Yim[arow1 * HH + ac], Yim[arow1 * HH + ac + 1]};
    v2f bRe0 = {Fre[jcol0 * HH + ac], Fre[jcol0 * HH + ac + 1]};  // B = F^T
    v2f bRe1 = {Fre[jcol1 * HH + ac], Fre[jcol1 * HH + ac + 1]};
    v2f bIm0 = {Fim[jcol0 * HH + ac], Fim[jcol0 * HH + ac + 1]};
    v2f bIm1 = {Fim[jcol1 * HH + ac], Fim[jcol1 * HH + ac + 1]};
    v2f bImN0 = {-bIm0[0], -bIm0[1]};  // f32 WMMA NEG is C-only -> VALU negate
    v2f bImN1 = {-bIm1[0], -bIm1[1]};
    zRe00 = wmma4(aRe0, bRe0, zRe00);
    zRe01 = wmma4(aRe0, bRe1, zRe01);
    zRe10 = wmma4(aRe1, bRe0, zRe10);
    zRe11 = wmma4(aRe1, bRe1, zRe11);
    zIm00 = wmma4(aRe0, bIm0, zIm00);
    zIm01 = wmma4(aRe0, bIm1, zIm01);
    zIm10 = wmma4(aRe1, bIm0, zIm10);
    zIm11 = wmma4(aRe1, bIm1, zIm11);
    zRe00 = wmma4(aIm0, bImN0, zRe00);
    zRe01 = wmma4(aIm0, bImN1, zRe01);
    zRe10 = wmma4(aIm1, bImN0, zRe10);
    zRe11 = wmma4(aIm1, bImN1, zRe11);
    zIm00 = wmma4(aIm0, bRe0, zIm00);
    zIm01 = wmma4(aIm0, bRe1, zIm01);
    zIm10 = wmma4(aIm1, bRe0, zIm10);
    zIm11 = wmma4(aIm1, bRe1, zIm11);
  }

#pragma unroll
  for (int r = 0; r < 8; ++r) {
    int k0 = kS * 32 + r + 8 * laneHi;
    int k1 = k0 + 16;
    int ks0 = (k0 + 128) & 255, ks1 = (k1 + 128) & 255;  // fftshift rows
    int js0 = (jcol0 + 128) & 255, js1 = (jcol1 + 128) & 255;
    float p;
    p = zRe00[r] * zRe00[r] + zIm00[r] * zIm00[r];
    atomicAdd(&hist[radial_bin(ks0, js0)], p);
    p = zRe01[r] * zRe01[r] + zIm01[r] * zIm01[r];
    atomicAdd(&hist[radial_bin(ks0, js1)], p);
    p = zRe10[r] * zRe10[r] + zIm10[r] * zIm10[r];
    atomicAdd(&hist[radial_bin(ks1, js0)], p);
    p = zRe11[r] * zRe11[r] + zIm11[r] * zIm11[r];
    atomicAdd(&hist[radial_bin(ks1, js1)], p);
  }
  __syncthreads();
  if (t < NBINS1) {
    int g = imgBase + imgLocal;
    atomicAdd(ws + OFF_BSUM + (size_t)g * NBINS1 + t, hist[t]);
  }
}

// --- stage 3: per-image log-log regression ---------------------------------
__global__ void __launch_bounds__(128) sl_slope(float* __restrict__ ws) {
  __shared__ float s0[128], s1[128];
  __shared__ float mlf, mlp;
  int g = blockIdx.x;
  int t = threadIdx.x;
  bool valid = (t >= 1);  // bins 1..127 (DC dropped)

  float lf = 0.0f, lp = 0.0f;
  if (valid) {
    const unsigned* cnt = (const unsigned*)(ws + OFF_CNT);
    float c = (float)cnt[t];
    float sum = ws[OFF_BSUM + (size_t)g * NBINS1 + t];
    float prof = sum / fmaxf(c, 1.0f);
    prof = (c > 0.0f) ? prof : 0.0f;
    lf = log10f(((float)t + 0.5f) * (1.0f / 256.0f) + 1e-10f);
    lp = log10f(prof + 1e-10f);
  }
  s0[t] = lf; s1[t] = lp;
  __syncthreads();
  for (int s = 64; s > 0; s >>= 1) {
    if (t < s) { s0[t] += s0[t + s]; s1[t] += s1[t + s]; }
    __syncthreads();
  }
  if (t == 0) { mlf = s0[0] * (1.0f / 127.0f); mlp = s1[0] * (1.0f / 127.0f); }
  __syncthreads();

  float xd = valid ? (lf - mlf) : 0.0f;
  float yd = valid ? (lp - mlp) : 0.0f;
  s0[t] = xd * xd; s1[t] = xd * yd;
  __syncthreads();
  for (int s = 64; s > 0; s >>= 1) {
    if (t < s) { s0[t] += s0[t + s]; s1[t] += s1[t + s]; }
    __syncthreads();
  }
  if (t == 0) {
    float slope = -s1[0] / s0[0];
    float loss = fabsf(slope - 2.0f) + fmaxf(1.5f - slope, 0.0f) +
                 fmaxf(slope - 2.5f, 0.0f);
    ws[OFF_LOSS + g] = loss;
  }
}

// --- stage 4: deterministic final reduction --------------------------------
__global__ void __launch_bounds__(512) sl_final(const float* __restrict__ ws,
                                                float* __restrict__ out) {
  __shared__ float s[512];
  int t = threadIdx.x;
  s[t] = ws[OFF_LOSS + t];
  __syncthreads();
  for (int st = 256; st > 0; st >>= 1) {
    if (t < st) s[t] += s[t + st];
    __syncthreads();
  }
  if (t == 0) out[0] = s[0] * (1.0f / (float)NIMG);
}

extern "C" void kernel_launch(void* const* d_in, const int* in_sizes, int n_in,
                              void* d_out, int out_size, void* d_ws,
                              size_t ws_size, hipStream_t stream) {
  (void)in_sizes; (void)n_in; (void)out_size; (void)ws_size;
  const float* x = (const float*)d_in[0];
  float* ws = (float*)d_ws;
  float* out = (float*)d_out;

  {
    int n = NIMG * NBINS1 + NBINS1;
    sl_init<<<(n + 255) / 256, 256, 0, stream>>>(ws);
  }
  sl_build<<<(HH * HH) / 256, 256, 0, stream>>>(ws);

  for (int chunk = 0; chunk < NCHUNK; ++chunk) {
    int imgBase = chunk * CHUNK;
    // 16 images * 64 super-tiles = 1024 waves -> 128 blocks of 8 waves
    sl_stage1<<<128, 256, 0, stream>>>(x, ws, imgBase);
    sl_stage2<<<128, 256, 0, stream>>>(ws, imgBase);
  }

  sl_slope<<<NIMG, 128, 0, stream>>>(ws);
  sl_final<<<1, 512, 0, stream>>>(ws, out);
}